// SemanticMemoryLatentSpace_70205535420689
// MI455X (gfx1250) — compile-verified
//
#include <hip/hip_runtime.h>
#include <math.h>

typedef __attribute__((ext_vector_type(2))) float v2f;
typedef __attribute__((ext_vector_type(8))) float v8f;

#define DIM    384
#define NBATCH 512
#define NCLUS  100
#define MOMF   0.95f
#define CAPF   50.0f
// ALPHA = 0.02 * sqrt(64/384)
#define ALPHAF 0.0081649658092772603f

// ---------------------------------------------------------------------------
// WMMA fragment conventions (wave32, V_WMMA_F32_16X16X4_F32):
//   A (16x4, MxK): lane<16 -> M=lane,    a[v]=A[M][v]   (K=0,1)
//                  lane>=16-> M=lane-16, a[v]=A[M][2+v] (K=2,3)
//   B (4x16, KxN): lane<16 -> N=lane,    b[v]=B[v][N]
//                  lane>=16-> N=lane-16, b[v]=B[2+v][N]
//   C/D (16x16):   lane<16 -> c[v]=C[v][lane]; lane>=16 -> c[v]=C[v+8][lane-16]
// ---------------------------------------------------------------------------

// S = E * E^T  (NBATCH x NBATCH), K-dim = DIM.  1024 tiles, 1 wave each.
__global__ __launch_bounds__(128) void gram_kernel(const float* __restrict__ E,
                                                   float* __restrict__ S) {
  const int lane = threadIdx.x & 31;
  const int wid  = blockIdx.x * 4 + (threadIdx.x >> 5);
  const int tm = wid >> 5, tn = wid & 31;           // 32 x 32 tiles
  const int half = lane >> 4, mn = lane & 15;
  const int rowA = tm * 16 + mn;                    // A: M index
  const int rowB = tn * 16 + mn;                    // B[k][n] = E[rowB][k]
  v8f acc = {};
  for (int k0 = 0; k0 < DIM; k0 += 4) {
    const int k = k0 + half * 2;
    v2f a, b;
    a.x = E[rowA * DIM + k];  a.y = E[rowA * DIM + k + 1];
    b.x = E[rowB * DIM + k];  b.y = E[rowB * DIM + k + 1];
    acc = __builtin_amdgcn_wmma_f32_16x16x4_f32(false, a, false, b,
                                                (short)0, acc, false, false);
  }
  const int col = tn * 16 + mn;
  #pragma unroll
  for (int v = 0; v < 8; ++v)
    S[(tm * 16 + v + half * 8) * NBATCH + col] = acc[v];
}

// M = E * K0  (NBATCH x DIM), K-dim = DIM.  768 tiles.
__global__ __launch_bounds__(128) void ek0_kernel(const float* __restrict__ E,
                                                  const float* __restrict__ K0,
                                                  float* __restrict__ M) {
  const int lane = threadIdx.x & 31;
  const int wid  = blockIdx.x * 4 + (threadIdx.x >> 5);
  const int tm = wid / 24, tn = wid % 24;           // 32 x 24 tiles
  const int half = lane >> 4, mn = lane & 15;
  const int rowA = tm * 16 + mn;
  const int colB = tn * 16 + mn;
  v8f acc = {};
  for (int k0 = 0; k0 < DIM; k0 += 4) {
    const int k = k0 + half * 2;
    v2f a, b;
    a.x = E[rowA * DIM + k];       a.y = E[rowA * DIM + k + 1];
    b.x = K0[k * DIM + colB];      b.y = K0[(k + 1) * DIM + colB];
    acc = __builtin_amdgcn_wmma_f32_16x16x4_f32(false, a, false, b,
                                                (short)0, acc, false, false);
  }
  #pragma unroll
  for (int v = 0; v < 8; ++v)
    M[(tm * 16 + v + half * 8) * DIM + colB] = acc[v];
}

// q[r] = dot(M[r,:], E[r,:])  -> e_r^T K0 e_r.  One wave per row.
__global__ __launch_bounds__(256) void rowdot_kernel(const float* __restrict__ M,
                                                     const float* __restrict__ E,
                                                     float* __restrict__ q) {
  const int lane = threadIdx.x & 31;
  const int row  = blockIdx.x * 8 + (threadIdx.x >> 5);
  float s = 0.f;
  for (int d = lane; d < DIM; d += 32) s += M[row * DIM + d] * E[row * DIM + d];
  for (int o = 16; o; o >>= 1) s += __shfl_down(s, o, 32);
  if (lane == 0) q[row] = s;
}

// Sequential scan: centroid bank lives in LDS; K reduced to scalar recurrences.
// Outputs g[j] (rank-1 coefficients, pre-scale) and final running scale R.
__global__ __launch_bounds__(1024) void scan_kernel(const float* __restrict__ E,
                                                    const float* __restrict__ K0,
                                                    const float* __restrict__ S,
                                                    const float* __restrict__ q,
                                                    float* __restrict__ g_out,
                                                    float* __restrict__ scal_out) {
  __shared__ float C[NCLUS * DIM];    // 153600 B centroid bank
  __shared__ float cn2[NCLUS];        // ||C_k||^2 tracked incrementally
  __shared__ float dotC[NCLUS];       // C_k . e_t (raw)
  __shared__ float ew[DIM];           // current embedding
  __shared__ float gsh[NBATCH];       // g_j coefficients
  __shared__ float red[32];           // per-wave partials
  __shared__ float bcast[4];          // [0]=prev_norm(||K||), [1]=R
  __shared__ int   ibcast[4];         // [0]=active, [1]=idx, [2]=create_new

  const int tid = threadIdx.x;
  const int lane = tid & 31, wave = tid >> 5;

  for (int i = tid; i < NCLUS * DIM; i += 1024) C[i] = 0.f;
  if (tid < NCLUS) cn2[tid] = 0.f;

  // ||K0||_F
  float ps = 0.f;
  for (int i = tid; i < DIM * DIM; i += 1024) { float v = K0[i]; ps += v * v; }
  for (int o = 16; o; o >>= 1) ps += __shfl_down(ps, o, 32);
  if (lane == 0) red[wave] = ps;
  __syncthreads();
  if (tid == 0) {
    float s = 0.f;
    for (int w = 0; w < 32; ++w) s += red[w];
    bcast[0] = sqrtf(s);   // prev_norm = ||K0||
    bcast[1] = 1.0f;       // R (running clip product)
    ibcast[0] = 0;         // active clusters
  }
  __syncthreads();

  for (int t = 0; t < NBATCH; ++t) {
    // stage e_t
    for (int i = tid; i < DIM; i += 1024) ew[i] = E[t * DIM + i];
    __syncthreads();

    // dotC[k] = C[k] . e_t   (8 threads per cluster, aligned sub-groups)
    if (tid < NCLUS * 8) {
      const int k = tid >> 3, c8 = tid & 7;
      float s = 0.f;
      const float* Ck = &C[k * DIM];
      for (int d = c8; d < DIM; d += 8) s += Ck[d] * ew[d];
      for (int o = 4; o; o >>= 1) s += __shfl_down(s, o, 32);
      if (c8 == 0) dotC[k] = s;
    }

    // partial of sum_{j<t} g_j * S[j,t]^2  (threads 0..511)
    float gs = 0.f;
    if (tid < t) { float sv = S[tid * NBATCH + t]; gs = gsh[tid] * sv * sv; }
    for (int o = 16; o; o >>= 1) gs += __shfl_down(gs, o, 32);
    if (lane == 0) red[wave] = gs;
    __syncthreads();

    if (wave == 0) {
      const int activ = ibcast[0];
      const float ee = S[t * NBATCH + t];        // ||e_t||^2
      const float enorm = sqrtf(ee);
      // max/argmax over active cluster cosine sims (first-max tie-break)
      float best = -2.f; int bidx = 0;
      for (int k = lane; k < NCLUS; k += 32) {
        if (k < activ) {
          float sim = dotC[k] / (sqrtf(cn2[k]) * enorm + 1e-8f);
          if (sim > best) { best = sim; bidx = k; }
        }
      }
      for (int o = 16; o; o >>= 1) {
        float ob = __shfl_down(best, o, 32);
        int   oi = __shfl_down(bidx, o, 32);
        if (ob > best || (ob == best && oi < bidx)) { best = ob; bidx = oi; }
      }
      if (lane == 0) {
        float ss = 0.f;
        for (int w = 0; w < 16; ++w) ss += red[w];
        const float prevnorm = bcast[0];
        const float R = bcast[1];
        float novelty = (activ == 0) ? 1.f : (1.f - best * best);
        novelty = fminf(fmaxf(novelty, 0.f), 1.f);
        const float iw = powf(novelty, 1.5f);
        // growth: tracked norm_before == carried prev_norm -> 0 -> plateau boost
        const float growth = 0.f;
        const float alph = (growth < 0.01f) ? (ALPHAF * 1.5f) : ALPHAF;
        float c = alph * iw;
        if (ee > 1e-8f) c *= (float)DIM / fmaxf(ee, 1e-8f);   // trace-normalized pert
        const float eKe = R * (q[t] + ss);                     // e^T K_{t-1} e
        const float kn2 = prevnorm * prevnorm + 2.f * c * eKe + c * c * ee * ee;
        const float kn = sqrtf(kn2);
        const float clip = (kn > CAPF) ? (CAPF / kn) : 1.f;
        gsh[t] = c / R;
        bcast[1] = R * clip;
        bcast[0] = kn * clip;
        // centroid bookkeeping
        const int create = (novelty > 0.7f) && (activ < NCLUS);
        const int idx = create ? activ : bidx;
        ibcast[1] = idx; ibcast[2] = create;
        if (create) ibcast[0] = activ + 1;
        if (create) cn2[idx] = ee;
        else cn2[idx] = MOMF * MOMF * cn2[idx]
                      + 2.f * MOMF * (1.f - MOMF) * dotC[idx]
                      + (1.f - MOMF) * (1.f - MOMF) * ee;
      }
    }
    __syncthreads();

    { // centroid vector update
      const int idx = ibcast[1], create = ibcast[2];
      for (int d = tid; d < DIM; d += 1024) {
        const float old = C[idx * DIM + d];
        C[idx * DIM + d] = create ? ew[d] : (MOMF * old + (1.f - MOMF) * ew[d]);
      }
    }
    __syncthreads();
  }

  for (int i = tid; i < NBATCH; i += 1024) g_out[i] = gsh[i];
  if (tid == 0) scal_out[0] = bcast[1];
}

// Kf = R * (K0 + E^T diag(g) E)   (DIM x DIM), K-dim = NBATCH.  576 tiles.
__global__ __launch_bounds__(128) void kfinal_kernel(const float* __restrict__ E,
                                                     const float* __restrict__ K0,
                                                     const float* __restrict__ g,
                                                     const float* __restrict__ scal,
                                                     float* __restrict__ Kf) {
  const int lane = threadIdx.x & 31;
  const int wid  = blockIdx.x * 4 + (threadIdx.x >> 5);
  const int tm = wid / 24, tn = wid % 24;           // 24 x 24 tiles
  const int half = lane >> 4, mn = lane & 15;
  const int rowO = tm * 16 + mn;                    // output row (A's M)
  const int colO = tn * 16 + mn;                    // output col (B's N)
  v8f acc;
  #pragma unroll
  for (int v = 0; v < 8; ++v)
    acc[v] = K0[(tm * 16 + v + half * 8) * DIM + tn * 16 + mn];
  for (int k0 = 0; k0 < NBATCH; k0 += 4) {
    const int j = k0 + half * 2;
    v2f a, b;
    a.x = E[j * DIM + rowO] * g[j];
    a.y = E[(j + 1) * DIM + rowO] * g[j + 1];
    b.x = E[j * DIM + colO];
    b.y = E[(j + 1) * DIM + colO];
    acc = __builtin_amdgcn_wmma_f32_16x16x4_f32(false, a, false, b,
                                                (short)0, acc, false, false);
  }
  const float R = scal[0];
  #pragma unroll
  for (int v = 0; v < 8; ++v)
    Kf[(tm * 16 + v + half * 8) * DIM + tn * 16 + mn] = R * acc[v];
}

// OUT = E * Kf  (NBATCH x DIM), K-dim = DIM.  768 tiles.
__global__ __launch_bounds__(128) void refined_kernel(const float* __restrict__ E,
                                                      const float* __restrict__ Kf,
                                                      float* __restrict__ OUT) {
  const int lane = threadIdx.x & 31;
  const int wid  = blockIdx.x * 4 + (threadIdx.x >> 5);
  const int tm = wid / 24, tn = wid % 24;           // 32 x 24 tiles
  const int half = lane >> 4, mn = lane & 15;
  const int rowA = tm * 16 + mn;
  const int colB = tn * 16 + mn;
  v8f acc = {};
  for (int k0 = 0; k0 < DIM; k0 += 4) {
    const int k = k0 + half * 2;
    v2f a, b;
    a.x = E[rowA * DIM + k];        a.y = E[rowA * DIM + k + 1];
    b.x = Kf[k * DIM + colB];       b.y = Kf[(k + 1) * DIM + colB];
    acc = __builtin_amdgcn_wmma_f32_16x16x4_f32(false, a, false, b,
                                                (short)0, acc, false, false);
  }
  #pragma unroll
  for (int v = 0; v < 8; ++v)
    OUT[(tm * 16 + v + half * 8) * DIM + colB] = acc[v];
}

extern "C" void kernel_launch(void* const* d_in, const int* in_sizes, int n_in,
                              void* d_out, int out_size, void* d_ws, size_t ws_size,
                              hipStream_t stream) {
  const float* E  = (const float*)d_in[0];   // [512, 384]
  const float* K0 = (const float*)d_in[1];   // [384, 384]
  float* out = (float*)d_out;                // [512, 384]
  float* w = (float*)d_ws;
  float* S    = w;                // 262144
  float* M    = w + 262144;       // 196608
  float* q    = w + 458752;       // 512
  float* g    = w + 459264;       // 512
  float* scal = w + 459776;       // 8
  float* Kf   = w + 460288;       // 147456

  gram_kernel  <<<256, 128, 0, stream>>>(E, S);
  ek0_kernel   <<<192, 128, 0, stream>>>(E, K0, M);
  rowdot_kernel<<< 64, 256, 0, stream>>>(M, E, q);
  scan_kernel  <<<  1, 1024, 0, stream>>>(E, K0, S, q, g, scal);
  kfinal_kernel<<<144, 128, 0, stream>>>(E, K0, g, scal, Kf);
  refined_kernel<<<192, 128, 0, stream>>>(E, Kf, out);
}